// MFCC_72954314490451
// MI455X (gfx1250) — compile-verified
//
#include <hip/hip_runtime.h>
#include <math.h>

// ---------------------------------------------------------------------------
// MFCC pipeline for MI455X (gfx1250, wave32).
// DFT + mel projection as fp32 GEMMs on matrix cores (V_WMMA_F32_16X16X4_F32,
// reference-accurate fp32 numerics). Basis tables are precomputed in the
// *WMMA B-fragment register order* so the hot loop issues a single
// global_load_b128 (cos+sin fragments) per K-step — fully coalesced,
// L2-resident. Hann window folded into the DFT basis.
// ---------------------------------------------------------------------------

typedef __attribute__((ext_vector_type(2))) float v2f;
typedef __attribute__((ext_vector_type(4))) float v4f;
typedef __attribute__((ext_vector_type(8))) float v8f;

#define BATCH  1024
#define NSAMP  16000
#define WINL   400
#define HOP    160
#define NFFT   512
#define NBINS  257
#define NMEL   40
#define NCEP   13
#define TFRM   98
#define MTILES 7        // ceil(98/16) frame tiles
#define NTILES 17       // 272 padded bins / 16
#define KS_DFT 100      // 400 / 4
#define KS_MEL 68       // 272 / 4
#define PI_F   3.14159265358979323846f

// workspace layout (floats); total = 230656 + 1024*98*13 = 1,535,232 (6.14 MB)
#define WS_DFTB 0                         // 17*100*32*4 = 217600 (cos|sin swizzled)
#define WS_MELB 217600                    // 3*68*32*2   = 13056  (mel swizzled)
#define WS_FEAT 230656                    // feats: 1024*98*13

// WMMA f32 16x16x4 fragment layout (ISA 7.12.2):
//   A (16x4):  lane L holds M = L&15;  vgpr v holds K = v + 2*(L>>4)
//   B (4x16):  lane L holds N = L&15;  vgpr v holds K = v + 2*(L>>4)
//   C/D:       lane L holds N = L&15;  vgpr v holds M = v + 8*(L>>4)

// ---------------------------------------------------------------------------
// Kernel 0: build swizzled DFT basis (window folded in) and mel matrix,
// laid out per (tile, kstep, lane) in B-fragment order.
// Exact phase range reduction: (k*b) mod 512 stays integer-exact.
// ---------------------------------------------------------------------------
__device__ __forceinline__ float dft_basis(int k, int n, int part /*0=cos,1=sin*/) {
  if (n >= NBINS) return 0.f;
  float w = 0.5f - 0.5f * cosf(2.f * PI_F * (float)k / (float)WINL); // Hann
  int m = (k * n) & (NFFT - 1);
  float ang = 2.f * PI_F * (float)m / (float)NFFT;
  return w * (part ? sinf(ang) : cosf(ang)); // sin sign irrelevant (|im|)
}

__global__ void mfcc_init(const float* __restrict__ mel_fb, float* __restrict__ ws) {
  int i = blockIdx.x * blockDim.x + threadIdx.x;
  if (i < NTILES * KS_DFT * 32) {                 // 54400 lane-slots, 4 floats each
    int tile = i / (KS_DFT * 32);
    int r    = i % (KS_DFT * 32);
    int ks   = r / 32, lane = r % 32;
    int lo = lane & 15, hi = lane >> 4;
    int n  = tile * 16 + lo;
    int k0 = ks * 4 + 2 * hi;
    float* o = ws + WS_DFTB + (size_t)i * 4;
    o[0] = dft_basis(k0,     n, 0);
    o[1] = dft_basis(k0 + 1, n, 0);
    o[2] = dft_basis(k0,     n, 1);
    o[3] = dft_basis(k0 + 1, n, 1);
  }
  if (i < 3 * KS_MEL * 32) {                      // 6528 lane-slots, 2 floats each
    int tile = i / (KS_MEL * 32);
    int r    = i % (KS_MEL * 32);
    int ks   = r / 32, lane = r % 32;
    int lo = lane & 15, hi = lane >> 4;
    int n  = tile * 16 + lo;
    int k0 = ks * 4 + 2 * hi;
    float* o = ws + WS_MELB + (size_t)i * 2;
    o[0] = (k0     < NBINS && n < NMEL) ? mel_fb[(k0    ) * NMEL + n] : 0.f;
    o[1] = (k0 + 1 < NBINS && n < NMEL) ? mel_fb[(k0 + 1) * NMEL + n] : 0.f;
  }
}

// ---------------------------------------------------------------------------
// Kernel 1: per (batch, 16-frame tile): DFT (WMMA f32), |.|, energy,
// mel GEMM (WMMA f32), log, DCT + lifter -> feats[b][t][13] in ws.
// ---------------------------------------------------------------------------
__global__ void __launch_bounds__(128) mfcc_frames(const float* __restrict__ wav,
                                                   float* __restrict__ ws) {
  __shared__ float fr[16 * 401];     // frames, stride 401 -> conflict-free columns
  __shared__ float spec[16 * 273];   // magnitudes, stride 273
  __shared__ float logmel[16 * 41];
  __shared__ float energy[16];
  __shared__ float dctT[12 * 40];

  const int tileM = blockIdx.x;
  const int b     = blockIdx.y;
  const int tid   = threadIdx.x;
  const int wave  = tid >> 5;        // 4 wave32s
  const int lane  = tid & 31;
  const int lo    = lane & 15;
  const int hi    = lane >> 4;

  const v4f* dftB = (const v4f*)(ws + WS_DFTB);  // [(tile*100+ks)*32 + lane]
  const v2f* melB = (const v2f*)(ws + WS_MELB);  // [(tile*68 +ks)*32 + lane]
  float*     feat = ws + WS_FEAT;

  if (tid < 16) energy[tid] = 0.f;
  for (int i = tid; i < 12 * 40; i += 128) {     // DCT-II rows 1..12
    int k = i / 40 + 1, j = i % 40;
    dctT[i] = 2.f * cosf(PI_F * (float)k * ((float)j + 0.5f) / (float)NMEL);
  }
  for (int i = tid; i < 16 * WINL; i += 128) {   // stage (zero-padded) frames
    int f = i / WINL, n = i % WINL;
    int t = tileM * 16 + f;
    fr[f * 401 + n] = (t < TFRM) ? wav[b * NSAMP + t * HOP + n] : 0.f;
  }
  __syncthreads();

  // ---- Phase 1: DFT. M=16 frames, K=400 samples, N=272 bins (cos & sin) ----
  for (int tile = wave; tile < NTILES; tile += 4) {
    v8f cacc = {}; v8f sacc = {};
    const int n = tile * 16 + lo;
    const v4f* bp = dftB + (size_t)tile * KS_DFT * 32 + lane;
    for (int ks = 0; ks < KS_DFT; ++ks) {
      const int k0 = ks * 4 + 2 * hi;
      v2f a;  a.x = fr[lo * 401 + k0];  a.y = fr[lo * 401 + k0 + 1];
      v4f t = bp[ks * 32];                      // one b128: cos & sin fragments
      v2f bc; bc.x = t.x; bc.y = t.y;
      v2f bs; bs.x = t.z; bs.y = t.w;
      cacc = __builtin_amdgcn_wmma_f32_16x16x4_f32(false, a, false, bc, (short)0, cacc, false, false);
      sacc = __builtin_amdgcn_wmma_f32_16x16x4_f32(false, a, false, bs, (short)0, sacc, false, false);
    }
    for (int v = 0; v < 8; ++v) {
      int m = v + 8 * hi;
      float re = cacc[v], im = sacc[v];
      spec[m * 273 + n] = (n < NBINS) ? sqrtf(re * re + im * im) : 0.f;
    }
  }
  __syncthreads();

  // ---- energy = sum(spec^2), partial per thread, ds_add_f32 combine ----
  {
    int f = tid & 15, c = tid >> 4;
    float s = 0.f;
    for (int nn = c; nn < NBINS; nn += 8) { float v = spec[f * 273 + nn]; s += v * v; }
    atomicAdd(&energy[f], s);
  }

  // ---- Phase 2: log-mel = log(spec @ mel + 1e-6). K=272, N=48 (3 tiles) ----
  if (wave < 3) {
    const int tile = wave;
    v8f acc = {};
    const v2f* mp = melB + (size_t)tile * KS_MEL * 32 + lane;
    for (int ks = 0; ks < KS_MEL; ++ks) {
      const int k0 = ks * 4 + 2 * hi;
      v2f a;  a.x = spec[lo * 273 + k0];  a.y = spec[lo * 273 + k0 + 1];
      v2f bm = mp[ks * 32];                     // one b64: mel fragment
      acc = __builtin_amdgcn_wmma_f32_16x16x4_f32(false, a, false, bm, (short)0, acc, false, false);
    }
    for (int v = 0; v < 8; ++v) {
      int m = v + 8 * hi, nn = tile * 16 + lo;
      if (nn < NMEL) logmel[m * 41 + nn] = logf(acc[v] + 1e-6f);
    }
  }
  __syncthreads();

  // ---- Phase 3: DCT (keep ceps 1..12), scale, lifter; + log-energy ----
  const float scale = 0.11180339887498949f;  // 1/sqrt(2*NMEL)
  for (int i = tid; i < 16 * 12; i += 128) {
    int f = i / 12, c = i % 12;               // c = cep index - 1
    int t = tileM * 16 + f;
    if (t < TFRM) {
      float s = 0.f;
      for (int j = 0; j < NMEL; ++j) s += logmel[f * 41 + j] * dctT[c * 40 + j];
      float lift = 1.f + 11.f * sinf(PI_F * (float)c / 22.f);
      feat[(b * TFRM + t) * NCEP + 1 + c] = s * scale * lift;
    }
  }
  if (tid < 16) {
    int t = tileM * 16 + tid;
    if (t < TFRM)
      feat[(b * TFRM + t) * NCEP] = logf(energy[tid] / (float)NFFT + 1e-6f);
  }
}

// ---------------------------------------------------------------------------
// Kernel 2: delta / delta-delta (N=2, replicated edges), assemble output.
// out[b][t][0:13]=feats, [13:26]=d1, [26:39]=d2; trailing dim 1 is flat.
// ---------------------------------------------------------------------------
__global__ void __launch_bounds__(128) mfcc_deltas(const float* __restrict__ ws_feat,
                                                   float* __restrict__ out) {
  __shared__ float fL[TFRM * NCEP];
  __shared__ float d1L[TFRM * NCEP];
  const int b = blockIdx.x, tid = threadIdx.x;

  for (int i = tid; i < TFRM * NCEP; i += 128) fL[i] = ws_feat[b * TFRM * NCEP + i];
  __syncthreads();

  for (int i = tid; i < TFRM * NCEP; i += 128) {
    int t = i / NCEP, c = i % NCEP;
    int tm2 = max(t - 2, 0), tm1 = max(t - 1, 0);
    int tp1 = min(t + 1, TFRM - 1), tp2 = min(t + 2, TFRM - 1);
    d1L[i] = (2.f * (fL[tp2 * NCEP + c] - fL[tm2 * NCEP + c]) +
                    (fL[tp1 * NCEP + c] - fL[tm1 * NCEP + c])) * 0.1f;
  }
  __syncthreads();

  for (int i = tid; i < TFRM * NCEP; i += 128) {
    int t = i / NCEP, c = i % NCEP;
    int tm2 = max(t - 2, 0), tm1 = max(t - 1, 0);
    int tp1 = min(t + 1, TFRM - 1), tp2 = min(t + 2, TFRM - 1);
    float d2 = (2.f * (d1L[tp2 * NCEP + c] - d1L[tm2 * NCEP + c]) +
                      (d1L[tp1 * NCEP + c] - d1L[tm1 * NCEP + c])) * 0.1f;
    float* o = out + ((size_t)(b * TFRM + t)) * 39;
    o[c]      = fL[i];
    o[13 + c] = d1L[i];
    o[26 + c] = d2;
  }
}

// ---------------------------------------------------------------------------
extern "C" void kernel_launch(void* const* d_in, const int* in_sizes, int n_in,
                              void* d_out, int out_size, void* d_ws, size_t ws_size,
                              hipStream_t stream) {
  (void)in_sizes; (void)n_in; (void)out_size; (void)ws_size;
  const float* wav = (const float*)d_in[0];   // (1024, 16000) f32
  const float* mel = (const float*)d_in[1];   // (257, 40) f32
  float* ws  = (float*)d_ws;                  // needs ~6.2 MB
  float* out = (float*)d_out;                 // (1024, 98, 39, 1) f32

  int initN = NTILES * KS_DFT * 32;           // 54400 slots covers both tables
  mfcc_init  <<<(initN + 255) / 256, 256, 0, stream>>>(mel, ws);
  mfcc_frames<<<dim3(MTILES, BATCH), 128, 0, stream>>>(wav, ws);
  mfcc_deltas<<<BATCH, 128, 0, stream>>>(ws + WS_FEAT, out);
}